// ExploratoryMechanism_24051816858306
// MI455X (gfx1250) — compile-verified
//
#include <hip/hip_runtime.h>

// Problem constants (from reference setup_inputs)
#define B_     4
#define S_     1024
#define C_     4096
#define KM_    64
#define D_     256
#define NCOL_  (C_ + KM_)   // 4160
#define TOPN_  16
#define NWAVE_ 4
#define NTILES_ (NCOL_ / 16) // 260

typedef __attribute__((ext_vector_type(2))) float v2f;
typedef __attribute__((ext_vector_type(8))) float v8f;
typedef __attribute__((ext_vector_type(4))) int v4i;

#if __has_builtin(__builtin_amdgcn_global_load_async_to_lds_b128)
#define HAVE_ASYNC_LDS 1
typedef __attribute__((address_space(1))) v4i gv4i;   // global int4
typedef __attribute__((address_space(3))) v4i lv4i;   // LDS int4
#else
#define HAVE_ASYNC_LDS 0
#endif

// ---------- helpers ----------

// Synchronous copy of a contiguous 16x256-float tile (16 KB) global->LDS, one wave.
__device__ __forceinline__ void copy_tile_wave(float* __restrict__ dst,
                                               const float* __restrict__ src,
                                               int lane) {
  const float4* s = (const float4*)src;
  float4* d = (float4*)dst;
#pragma unroll
  for (int i = 0; i < 32; ++i)          // 1024 float4 total / 32 lanes
    d[lane + 32 * i] = s[lane + 32 * i];
}

// Async (DMA) staging of the same tile: bypasses VGPRs, tracked by ASYNCcnt.
__device__ __forceinline__ void tile_prefetch(float* __restrict__ dst,
                                              const float* __restrict__ src,
                                              int lane) {
#if HAVE_ASYNC_LDS
#pragma unroll
  for (int i = 0; i < 32; ++i) {
    const int e = (lane + 32 * i) * 4;  // float index, 16B chunks
    __builtin_amdgcn_global_load_async_to_lds_b128(
        (gv4i*)(src + e), (lv4i*)(dst + e), 0, 0);
  }
#else
  copy_tile_wave(dst, src, lane);
#endif
}

// Wait for outstanding async tile DMAs (or close the sync-copy DS RAW).
__device__ __forceinline__ void tile_wait() {
#if HAVE_ASYNC_LDS
#if __has_builtin(__builtin_amdgcn_s_wait_asynccnt)
  __builtin_amdgcn_s_wait_asynccnt(0);
  asm volatile("" ::: "memory");
#else
  asm volatile("s_wait_asynccnt 0x0" ::: "memory");
#endif
#else
  asm volatile("s_wait_dscnt 0x0" ::: "memory");
#endif
}

// Copy with all threads of the block.
__device__ __forceinline__ void copy_tile_block(float* __restrict__ dst,
                                                const float* __restrict__ src,
                                                int tid, int nthreads) {
  const float4* s = (const float4*)src;
  float4* d = (float4*)dst;
  for (int i = tid; i < (16 * D_) / 4; i += nthreads)
    d[i] = s[i];
}

// 16x16 f32 dot-product tile over K=256 using V_WMMA_F32_16X16X4_F32.
// lda: 16 rows (M) x 256 (K) row-major in LDS ; ldb: 16 rows (N) x 256 (K).
// A layout (16x4 f32): lanes 0-15 M=0..15; VGPR0 = K {0 | 2}, VGPR1 = K {1 | 3}
// by lane half -> per lane load K = k + 2*half, k + 2*half + 1 (8B aligned).
__device__ __forceinline__ v8f wmma_dot_16x16(const float* __restrict__ lda,
                                              const float* __restrict__ ldb,
                                              int lane) {
  v8f acc = {};
  const int half2 = (lane >> 4) << 1;   // 0 or 2
  const int lr = lane & 15;
#pragma unroll 8
  for (int k = 0; k < D_; k += 4) {
    v2f a = *(const v2f*)(lda + lr * D_ + k + half2);
    v2f b = *(const v2f*)(ldb + lr * D_ + k + half2);
    acc = __builtin_amdgcn_wmma_f32_16x16x4_f32(
        /*neg_a=*/false, a, /*neg_b=*/false, b,
        /*c_mod=*/(short)0, acc, /*reuse_a=*/false, /*reuse_b=*/false);
  }
  return acc;
}

// wave-local LDS ordering: LDS ops are in-order per wave; this closes the
// cross-lane RAW (store->load) without a block-wide barrier.
__device__ __forceinline__ void wave_lds_fence() {
  asm volatile("s_wait_dscnt 0x0" ::: "memory");
}

// ---------- kernel 1: qp = q @ W^T + b ----------
__global__ void __launch_bounds__(32)
project_kernel(const float* __restrict__ q, const float* __restrict__ W,
               const float* __restrict__ bias, float* __restrict__ qp) {
  __shared__ float lq[16 * D_];
  __shared__ float lw[16 * D_];
  const int tile = blockIdx.x;          // B * (S/16) * (D/16) = 4096
  const int et = tile & 15;             // D/16 = 16
  const int st = (tile >> 4) & 63;      // S/16 = 64
  const int b  = tile >> 10;
  const int lane = threadIdx.x;

  tile_prefetch(lq, q + ((size_t)(b * S_) + st * 16) * D_, lane);
  tile_prefetch(lw, W + (size_t)et * 16 * D_, lane);  // rows e0..e0+15 of W
  tile_wait();
  wave_lds_fence();

  v8f acc = wmma_dot_16x16(lq, lw, lane);

  // C/D layout: col N = lane&15, row M = r + 8*(lane>>4)
  const int n = lane & 15, half = lane >> 4;
  const float bv = bias[et * 16 + n];
  float* outp = qp + ((size_t)(b * S_) + st * 16) * D_ + et * 16 + n;
#pragma unroll
  for (int r = 0; r < 8; ++r)
    outp[(size_t)(r + 8 * half) * D_] = acc[r] + bv;
}

// ---------- kernel 2: column squared norms ----------
__global__ void __launch_bounds__(256)
norms_kernel(const float* __restrict__ ctx, const float* __restrict__ mem,
             float* __restrict__ cn) {
  const int i = blockIdx.x * 256 + threadIdx.x;
  if (i >= B_ * NCOL_) return;
  const int b = i / NCOL_;
  const int col = i - b * NCOL_;
  const float* v = (col < C_) ? ctx + ((size_t)(b * C_) + col) * D_
                              : mem + ((size_t)(b * KM_) + (col - C_)) * D_;
  float s = 0.f;
#pragma unroll 8
  for (int k = 0; k < D_; k += 4) {
    float4 x = *(const float4*)&v[k];
    s += x.x * x.x + x.y * x.y + x.z * x.z + x.w * x.w;
  }
  cn[i] = s;
}

// column-tile source pointer (context for col<4096, memory after)
__device__ __forceinline__ const float* col_src(const float* __restrict__ ctx,
                                                const float* __restrict__ mem,
                                                int b, int col0) {
  return (col0 < C_) ? ctx + ((size_t)(b * C_) + col0) * D_
                     : mem + ((size_t)(b * KM_) + (col0 - C_)) * D_;
}

// ---------- kernel 3: fused distances + top-16 ----------
__global__ void __launch_bounds__(128)
dist_topk_kernel(const float* __restrict__ qp, const float* __restrict__ ctx,
                 const float* __restrict__ mem, const float* __restrict__ cn,
                 float* __restrict__ outv, int* __restrict__ outi) {
  __shared__ float la[16 * D_];                 // qp tile, reused 260x
  __shared__ float lb[NWAVE_][2][16 * D_];      // per-wave double-buffered tile
  __shared__ float ldist[NWAVE_][16 * 17];      // padded transpose buffer
  __shared__ float lqn[16];
  __shared__ float poolv[16][NWAVE_ * TOPN_];
  __shared__ int   pooli[16][NWAVE_ * TOPN_];

  const int tid = threadIdx.x;
  const int lane = tid & 31;
  const int w = tid >> 5;                       // wave id 0..3
  const int blk = blockIdx.x;                   // B * S/16 = 256
  const int st = blk & 63;
  const int b = blk >> 6;

  copy_tile_block(la, qp + ((size_t)(b * S_) + st * 16) * D_, tid, 128);
  __syncthreads();

  if (tid < 16) {
    float s = 0.f;
#pragma unroll 8
    for (int k = 0; k < D_; k += 4) {
      float4 x = *(const float4*)&la[tid * D_ + k];
      s += x.x * x.x + x.y * x.y + x.z * x.z + x.w * x.w;
    }
    lqn[tid] = s;
  }
  __syncthreads();

  // per-lane sorted ascending top-16 (lanes 0-15 own rows 0-15)
  float tv[TOPN_];
  int ti[TOPN_];
#pragma unroll
  for (int t = 0; t < TOPN_; ++t) { tv[t] = 3.4e38f; ti[t] = 0; }

  // prime the pipeline: async-prefetch this wave's first tile
  tile_prefetch(lb[w][0], col_src(ctx, mem, b, w * 16), lane);

  int cur = 0;
  for (int ct = w; ct < NTILES_; ct += NWAVE_) {   // 65 tiles per wave
    const int col0 = ct * 16;

    tile_wait();                                   // lb[w][cur] ready
    const int nt = ct + NWAVE_;
    if (nt < NTILES_)                              // overlap DMA with compute
      tile_prefetch(lb[w][cur ^ 1], col_src(ctx, mem, b, nt * 16), lane);

    v8f acc = wmma_dot_16x16(la, lb[w][cur], lane);
    cur ^= 1;

    const int n = lane & 15, half = lane >> 4;
    const float cnv = cn[b * NCOL_ + col0 + n];
#pragma unroll
    for (int r = 0; r < 8; ++r) {
      const int m = r + 8 * half;
      const float d2 = lqn[m] + cnv - 2.f * acc[r];
      ldist[w][m * 17 + n] = sqrtf(fmaxf(d2, 0.f));
    }
    wave_lds_fence();

    if (lane < 16) {
#pragma unroll 1
      for (int j = 0; j < 16; ++j) {
        float v = ldist[w][lane * 17 + j];
        if (v < tv[TOPN_ - 1]) {                 // early reject
          int gi = col0 + j;
#pragma unroll
          for (int t = 0; t < TOPN_; ++t) {      // bubble insert (ascending)
            if (v < tv[t]) {
              float fv = tv[t]; int fi = ti[t];
              tv[t] = v; ti[t] = gi;
              v = fv; gi = fi;
            }
          }
        }
      }
    }
    wave_lds_fence();   // protect ldist reuse next iteration
  }

  // cross-wave merge: 4 sorted 16-lists per row -> final top-16
  if (lane < 16) {
#pragma unroll
    for (int t = 0; t < TOPN_; ++t) {
      poolv[lane][w * TOPN_ + t] = tv[t];
      pooli[lane][w * TOPN_ + t] = ti[t];
    }
  }
  __syncthreads();

  if (tid < 16) {
    float fv[TOPN_];
    int fi[TOPN_];
#pragma unroll
    for (int t = 0; t < TOPN_; ++t) { fv[t] = 3.4e38f; fi[t] = 0; }
#pragma unroll 1
    for (int j = 0; j < NWAVE_ * TOPN_; ++j) {
      float v = poolv[tid][j];
      if (v < fv[TOPN_ - 1]) {
        int gi = pooli[tid][j];
#pragma unroll
        for (int t = 0; t < TOPN_; ++t) {
          if (v < fv[t]) {
            float xv = fv[t]; int xi = fi[t];
            fv[t] = v; fi[t] = gi;
            v = xv; gi = xi;
          }
        }
      }
    }
    const size_t base = ((size_t)(b * S_) + st * 16 + tid) * TOPN_;
#pragma unroll
    for (int t = 0; t < TOPN_; ++t) {
      outv[base + t] = fv[t];
      outi[base + t] = fi[t];
    }
  }
}

// ---------- launch ----------
extern "C" void kernel_launch(void* const* d_in, const int* in_sizes, int n_in,
                              void* d_out, int out_size, void* d_ws, size_t ws_size,
                              hipStream_t stream) {
  const float* q    = (const float*)d_in[0];  // [B,S,D]
  const float* ctx  = (const float*)d_in[1];  // [B,C,D]
  const float* mem  = (const float*)d_in[2];  // [B,K,D]
  const float* W    = (const float*)d_in[3];  // [D,D]
  const float* bias = (const float*)d_in[4];  // [D]

  float* outv = (float*)d_out;                           // [B,S,16] distances
  int*   outi = (int*)d_out + (size_t)B_ * S_ * TOPN_;   // [B,S,16] indices

  float* qp = (float*)d_ws;                   // B*S*D floats = 4 MB
  float* cn = qp + (size_t)B_ * S_ * D_;      // B*NCOL floats

  project_kernel<<<B_ * (S_ / 16) * (D_ / 16), 32, 0, stream>>>(q, W, bias, qp);
  norms_kernel<<<(B_ * NCOL_ + 255) / 256, 256, 0, stream>>>(ctx, mem, cn);
  dist_topk_kernel<<<B_ * (S_ / 16), 128, 0, stream>>>(qp, ctx, mem, cn, outv, outi);
}